// S2ConvNet_ForDepthData_670014898820
// MI455X (gfx1250) — compile-verified
//
#include <hip/hip_runtime.h>
#include <math.h>

#define PI_F 3.14159265358979323846f

#define BT   128
#define NIN  60
#define N1   20
#define N2   12
#define F1c  20
#define F2c  40
#define F2P  48     // padded F2 for divergence-free WMMA tiles (3 x 16)
#define KT1  100
#define KT2  286
#define G1c  24
#define G3c  192

typedef __attribute__((ext_vector_type(2))) float v2f;
typedef __attribute__((ext_vector_type(8))) float v8f;
typedef __attribute__((ext_vector_type(4))) int v4i;

#if defined(__HIP_DEVICE_COMPILE__) && __has_builtin(__builtin_amdgcn_global_load_async_to_lds_b128) && __has_builtin(__builtin_amdgcn_s_wait_asynccnt)
#define USE_ASYNC_LDS 1
typedef __attribute__((address_space(1))) v4i g_v4i;   // global int4
typedef __attribute__((address_space(3))) v4i l_v4i;   // LDS int4
#else
#define USE_ASYNC_LDS 0
#endif

__device__ __forceinline__ int modp(int x, int m) { int r = x % m; return r < 0 ? r + m : r; }

__device__ __forceinline__ int sqoff(int l) {
  int s = 0;
  for (int i = 0; i < l; ++i) { int L = 2 * i + 1; s += L * L; }
  return s;
}

// Driscoll-Healy quadrature weight j for bandwidth bw
__device__ __forceinline__ float qw_dev(int bw, int j) {
  float theta = PI_F * (float)(2 * j + 1) / (4.0f * (float)bw);
  float s = 0.f;
  for (int k = 0; k < bw; ++k) s += sinf(theta * (float)(2 * k + 1)) / (float)(2 * k + 1);
  return (2.0f / (float)bw) * sinf(theta) * s;
}

// c_i = sqrt(l(l+1) - m(m+1)), m = i - l  (superdiagonal of Jp)
__device__ __forceinline__ float csup(int l, int i) {
  float m = (float)(i - l);
  return sqrtf((float)(l * (l + 1)) - m * (m + 1.0f));
}

// Compute d^l(beta) = exp(beta * A), A = 0.5(Jp^T - Jp), into LDS D (n*n, n=2l+1).
// Scaling (2^-8) + Taylor(8) + 8 squarings. P0/P1 are LDS scratch (361 floats each).
__device__ void wigner_exp(int l, float beta, float* D, float* P0, float* P1, int tid, int nth) {
  int n = 2 * l + 1;
  float scale = beta * (1.0f / 256.0f);
  for (int e = tid; e < n * n; e += nth) {
    int r = e / n, c = e % n;
    float bs = 0.f;
    if (c == r + 1) bs = 0.5f * scale * csup(l, r);
    else if (c == r - 1) bs = -0.5f * scale * csup(l, c);
    P0[e] = bs;
    D[e] = (r == c ? 1.0f : 0.0f) + bs;
  }
  __syncthreads();
  float* Pa = P0; float* Pb = P1;
  for (int t = 2; t <= 8; ++t) {
    float inv = 1.0f / (float)t;
    for (int e = tid; e < n * n; e += nth) {
      int r = e / n, c = e % n;
      float v = 0.f;
      if (c > 0)     v += Pa[r * n + c - 1] * (0.5f * scale * csup(l, c - 1));
      if (c < n - 1) v += Pa[r * n + c + 1] * (-0.5f * scale * csup(l, c));
      Pb[e] = v * inv;
    }
    __syncthreads();
    for (int e = tid; e < n * n; e += nth) D[e] += Pb[e];
    __syncthreads();
    float* tmp = Pa; Pa = Pb; Pb = tmp;
  }
  for (int s = 0; s < 8; ++s) {
    for (int e = tid; e < n * n; e += nth) {
      int r = e / n, c = e % n;
      float v = 0.f;
      for (int k = 0; k < n; ++k) v += D[r * n + k] * D[k * n + c];
      Pa[e] = v;
    }
    __syncthreads();
    for (int e = tid; e < n * n; e += nth) D[e] = Pa[e];
    __syncthreads();
  }
}

// ---------------- setup kernels ----------------

__global__ __launch_bounds__(256) void k_dm1(float* DM1, int* MI1) {
  __shared__ float D[361], P0[361], P1[361];
  int j = blockIdx.x, l = blockIdx.y, tid = threadIdx.x;
  float beta = PI_F * (float)(2 * j + 1) / 120.0f;
  wigner_exp(l, beta, D, P0, P1, tid, 256);
  int L = 2 * l + 1, off = l * l;
  float wj = qw_dev(30, j);
  for (int r = tid; r < L; r += 256)
    DM1[(off + r) * NIN + j] = D[r * L + l] * wj;
  if (j == 0)
    for (int r = tid; r < L; r += 256)
      MI1[off + r] = modp(r - l, NIN);
}

__global__ __launch_bounds__(256) void k_y1(float* Y1r, float* Y1i) {
  __shared__ float D[361], P0[361], P1[361];
  int bidx = blockIdx.x, l = blockIdx.y, tid = threadIdx.x;
  float beta = (float)(bidx + 1) * (PI_F / 8.0f) / 3.0f;
  wigner_exp(l, beta, D, P0, P1, tid, 256);
  int L = 2 * l + 1, off = l * l;
  for (int e = tid; e < L * 8; e += 256) {
    int r = e / 8, a = e % 8;
    float alpha = (PI_F / 4.0f) * (float)a;
    float ang = (float)(r - l) * alpha;
    float dv = D[r * L + l];
    int g = bidx * 8 + a;
    Y1r[(off + r) * G1c + g] = dv * cosf(ang);
    Y1i[(off + r) * G1c + g] = -dv * sinf(ang);
  }
}

__global__ __launch_bounds__(256) void k_db1(float* DB1, float* DM2, int* MI2, int* NI2) {
  __shared__ float D[361], P0[361], P1[361];
  int j = blockIdx.x, l = blockIdx.y, tid = threadIdx.x;
  float beta = PI_F * (float)(2 * j + 1) / 40.0f;
  wigner_exp(l, beta, D, P0, P1, tid, 256);
  int L = 2 * l + 1;
  int base1 = 20 * sqoff(l) + j * L * L;
  for (int e = tid; e < L * L; e += 256) DB1[base1 + e] = D[e];
  if (l < 6) {
    float wj = qw_dev(10, j);
    int off2 = sqoff(l);
    for (int e = tid; e < L * L; e += 256)
      DM2[(off2 + e) * N1 + j] = D[e] * wj;
    if (j == 0)
      for (int e = tid; e < L * L; e += 256) {
        int m = e / L, n = e % L;
        MI2[off2 + e] = modp(m - l, N1);
        NI2[off2 + e] = modp(n - l, N1);
      }
  }
}

__global__ __launch_bounds__(256) void k_db2(float* DB2, float* W2J) {
  __shared__ float D[361], P0[361], P1[361];
  int j = blockIdx.x, l = blockIdx.y, tid = threadIdx.x;
  float beta = PI_F * (float)(2 * j + 1) / 24.0f;
  wigner_exp(l, beta, D, P0, P1, tid, 256);
  int L = 2 * l + 1;
  int base = 12 * sqoff(l) + j * L * L;
  for (int e = tid; e < L * L; e += 256) DB2[base + e] = D[e];
  if (l == 0 && tid == 0) W2J[j] = qw_dev(6, j);
}

__global__ __launch_bounds__(256) void k_y2(float* Y2r, float* Y2i) {
  __shared__ float D[361], P0[361], P1[361];
  int bidx = blockIdx.x, l = blockIdx.y, tid = threadIdx.x;
  float beta = (float)(bidx + 1) * (PI_F / 8.0f) / 3.0f;
  wigner_exp(l, beta, D, P0, P1, tid, 256);
  int L = 2 * l + 1, off2 = sqoff(l);
  for (int e = tid; e < L * L * 64; e += 256) {
    int mn = e >> 6, ac = e & 63;
    int a = ac >> 3, c = ac & 7;
    int m = mn / L - l, n = mn % L - l;
    float alpha = (PI_F / 4.0f) * (float)a;
    float pg = -6.283185307179586f + (float)c * (PI_F / 2.0f);
    float gamma = pg - alpha;
    float ang = (float)m * alpha + (float)n * gamma;
    float dv = D[mn];
    int g = (bidx * 8 + a) * 8 + c;
    Y2r[(off2 + mn) * G3c + g] = dv * cosf(ang);
    Y2i[(off2 + mn) * G3c + g] = -dv * sinf(ang);
  }
}

// ---------------- main pipeline ----------------

// 60-point row DFT: xh[b,j,m] = (1/60) sum_p x[b,0,j,p] e^{-2pi i m p/60}
__global__ __launch_bounds__(64) void k_fft1(const float* __restrict__ x, float* XHr, float* XHi) {
  int b = blockIdx.x, j = blockIdx.y, m = threadIdx.x;
  if (m < NIN) {
    const float* row = x + (size_t)b * NIN * NIN + (size_t)j * NIN;
    float sr = 0.f, si = 0.f;
    for (int p = 0; p < NIN; ++p) {
      float sv, cv; sincosf((2.0f * PI_F / NIN) * (float)(m * p), &sv, &cv);
      float xv = row[p];
      sr += xv * cv; si -= xv * sv;
    }
    size_t o = (size_t)b * NIN * NIN + (size_t)j * NIN + m;
    XHr[o] = sr * (1.0f / NIN); XHi[o] = si * (1.0f / NIN);
  }
}

// X[k,b] = sum_j DM1[k,j] * xh[b,j,MI1[k]]
__global__ __launch_bounds__(128) void k_xproj(const float* __restrict__ DM1, const int* __restrict__ MI1,
                                               const float* __restrict__ XHr, const float* __restrict__ XHi,
                                               float* Xr, float* Xi) {
  int k = blockIdx.x, b = threadIdx.x;
  int mi = MI1[k];
  float sr = 0.f, si = 0.f;
  for (int j = 0; j < NIN; ++j) {
    float d = DM1[k * NIN + j];
    size_t o = (size_t)b * NIN * NIN + (size_t)j * NIN + mi;
    sr += d * XHr[o]; si += d * XHi[o];
  }
  Xr[k * BT + b] = sr; Xi[k * BT + b] = si;
}

// Yk1[k,o] = SC1 * sum_g Y1[k,g]*kernel1[0,o,g]
__global__ __launch_bounds__(32) void k_yk1(const float* __restrict__ Y1r, const float* __restrict__ Y1i,
                                            const float* __restrict__ kernel1, float* YK1r, float* YK1i) {
  int k = blockIdx.x, o = threadIdx.x;
  if (o < F1c) {
    float sr = 0.f, si = 0.f;
    for (int g = 0; g < G1c; ++g) {
      float kv = kernel1[o * G1c + g];
      sr += Y1r[k * G1c + g] * kv;
      si += Y1i[k * G1c + g] * kv;
    }
    const float SC1 = 0.20412414523193154f; // 1/sqrt(24)
    YK1r[k * F1c + o] = sr * SC1; YK1i[k * F1c + o] = si * SC1;
  }
}

// Stage-1 synthesis (20x20 ifft2), ReLU, analysis at the 11x11 needed freqs,
// projection with DM2 -> X2[(k*20+i), b]  (A-matrix-friendly layout). One block per (b,i).
__global__ __launch_bounds__(256) void k_stage1(const float* __restrict__ DB1, const float* __restrict__ DM2,
                                                const int* __restrict__ MI2, const int* __restrict__ NI2,
                                                const float* __restrict__ Xr, const float* __restrict__ Xi,
                                                const float* __restrict__ YK1r, const float* __restrict__ YK1i,
                                                float* X2r, float* X2i) {
  __shared__ float Sr[400], Si[400], TRr[400], TRi[400], x1t[400];
  __shared__ float Ur[220], Ui[220], Gr[121], Gi[121];
  __shared__ float accr[KT2], acci[KT2];
  const int b = blockIdx.x, i = blockIdx.y, tid = threadIdx.x;
  const int nth = 256;
  for (int k = tid; k < KT2; k += nth) { accr[k] = 0.f; acci[k] = 0.f; }
  __syncthreads();
  for (int j = 0; j < N1; ++j) {
    for (int e = tid; e < 400; e += nth) { Sr[e] = 0.f; Si[e] = 0.f; }
    __syncthreads();
    for (int l = 0; l < 10; ++l) {
      int L = 2 * l + 1, off = l * l;
      int base = 20 * sqoff(l) + j * L * L;
      float fl = (float)(2 * l + 1);
      for (int e = tid; e < L * L; e += nth) {
        int m = e / L, n = e % L;
        float d = DB1[base + e] * fl;
        float xr = Xr[(off + m) * BT + b], xi = Xi[(off + m) * BT + b];
        float yr = YK1r[(off + n) * F1c + i], yi = YK1i[(off + n) * F1c + i];
        int mi = modp(m - l, N1), ni = modp(n - l, N1);
        Sr[mi * N1 + ni] += d * (xr * yr + xi * yi);
        Si[mi * N1 + ni] += d * (xi * yr - xr * yi);
      }
      __syncthreads();
    }
    for (int e = tid; e < 400; e += nth) {
      int m = e / N1, q = e % N1;
      float sr = 0.f, si = 0.f;
      for (int n = 0; n < N1; ++n) {
        float sv, cv; sincosf((2.0f * PI_F / N1) * (float)(n * q), &sv, &cv);
        float a = Sr[m * N1 + n], bb = Si[m * N1 + n];
        sr += a * cv - bb * sv;
        si += a * sv + bb * cv;
      }
      TRr[e] = sr; TRi[e] = si;
    }
    __syncthreads();
    for (int e = tid; e < 400; e += nth) {
      int pp = e / N1, q = e % N1;
      float v = 0.f;
      for (int m = 0; m < N1; ++m) {
        float sv, cv; sincosf((2.0f * PI_F / N1) * (float)(m * pp), &sv, &cv);
        v += TRr[m * N1 + q] * cv - TRi[m * N1 + q] * sv;
      }
      x1t[e] = fmaxf(v, 0.f);
    }
    __syncthreads();
    for (int e = tid; e < 11 * N1; e += nth) {
      int t = e / N1, q = e % N1;
      int v = (t < 6) ? t : t + 9;
      float sr = 0.f, si = 0.f;
      for (int pp = 0; pp < N1; ++pp) {
        float sv, cv; sincosf((2.0f * PI_F / N1) * (float)(v * pp), &sv, &cv);
        float xv = x1t[pp * N1 + q];
        sr += xv * cv; si -= xv * sv;
      }
      Ur[e] = sr; Ui[e] = si;
    }
    __syncthreads();
    for (int e = tid; e < 121; e += nth) {
      int t = e / 11, u = e % 11;
      int vu = (u < 6) ? u : u + 9;
      float sr = 0.f, si = 0.f;
      for (int q = 0; q < N1; ++q) {
        float sv, cv; sincosf((2.0f * PI_F / N1) * (float)(vu * q), &sv, &cv);
        float a = Ur[t * N1 + q], bb = Ui[t * N1 + q];
        sr += a * cv + bb * sv;
        si += bb * cv - a * sv;
      }
      Gr[e] = sr * (1.0f / 400.0f);
      Gi[e] = si * (1.0f / 400.0f);
    }
    __syncthreads();
    for (int k = tid; k < KT2; k += nth) {
      int v1 = MI2[k], v2 = NI2[k];
      int t1 = (v1 < 8) ? v1 : v1 - 9;
      int t2 = (v2 < 8) ? v2 : v2 - 9;
      float dm = DM2[k * N1 + j];
      accr[k] += dm * Gr[t1 * 11 + t2];
      acci[k] += dm * Gi[t1 * 11 + t2];
    }
    __syncthreads();
  }
  // X2 layout: [(k*F1c + i)][b]  -> GEMM A address is affine in the K index
  for (int k = tid; k < KT2; k += nth) {
    size_t o = ((size_t)k * F1c + i) * BT + b;
    X2r[o] = accr[k]; X2i[o] = acci[k];
  }
}

// Yk2[k,i,o] = SC2 * sum_g Y2[k,g]*kernel2[i,o,g], padded to F2P columns (zeros)
__global__ __launch_bounds__(256) void k_yk2(const float* __restrict__ Y2r, const float* __restrict__ Y2i,
                                             const float* __restrict__ kernel2, float* YK2r, float* YK2i) {
  int k = blockIdx.x, tid = threadIdx.x;
  const float SC2 = 0.016137430609197569f; // 1/sqrt(192*20)
  for (int e = tid; e < F1c * F2P; e += 256) {
    int i = e / F2P, o = e % F2P;
    float sr = 0.f, si = 0.f;
    if (o < F2c) {
      for (int g = 0; g < G3c; ++g) {
        float kv = kernel2[(i * F2c + o) * G3c + g];
        sr += Y2r[k * G3c + g] * kv;
        si += Y2i[k * G3c + g] * kv;
      }
      sr *= SC2; si *= SC2;
    }
    YK2r[(size_t)k * (F1c * F2P) + e] = sr;
    YK2i[(size_t)k * (F1c * F2P) + e] = si;
  }
}

// Per (l,m,n) pair: complex GEMM  Z[b,o] = sum_{k',i} X2[(m,k'),b,i] * conj(Yk2[(n,k'),i,o])
// M=128 (b), N=48 padded (o), K=L*20.  One wave32 per 16x16 tile, V_WMMA_F32_16X16X4_F32.
// B panel staged to LDS with gfx1250 async global->LDS loads when available.
__global__ __launch_bounds__(32) void k_zgemm(const float* __restrict__ X2r, const float* __restrict__ X2i,
                                              const float* __restrict__ Y2r, const float* __restrict__ Y2i,
                                              float* __restrict__ Zr, float* __restrict__ Zi) {
  __shared__ __align__(16) float BPr[220 * 16];
  __shared__ __align__(16) float BPi[220 * 16];
  int p = blockIdx.x;
  int l = 0, off = 0;
  for (;;) { int L = 2 * l + 1; if (p < off + L * L) break; off += L * L; ++l; }
  int L = 2 * l + 1;
  int rem = p - off;
  int mm = rem / L, nn = rem % L;
  int KK = L * F1c;
  int lane = threadIdx.x;
  int bRow0 = blockIdx.y * 16;
  int oCol0 = blockIdx.z * 16;
  int arow = bRow0 + (lane & 15);
  int ncol = lane & 15;
  int ocol = oCol0 + ncol;
  int khalf = (lane >> 4) * 2;
  const size_t aBase = (size_t)(off + mm * L) * (F1c * BT);
  const size_t bBase = (size_t)(off + nn * L) * (F1c * F2P);
  const float* Ar0 = X2r + aBase + arow;   // + kk*BT
  const float* Ai0 = X2i + aBase + arow;
  const float* Br0 = Y2r + bBase + oCol0;  // + kk*F2P (+col)
  const float* Bi0 = Y2i + bBase + oCol0;
  __builtin_prefetch(Ar0, 0, 1);
  __builtin_prefetch(Br0, 0, 1);

#if USE_ASYNC_LDS
  // Stage the B panel (KK x 16, real+imag) into LDS asynchronously (b128 = 4 floats).
  for (int e = lane * 4; e < KK * 16; e += 32 * 4) {
    int kk = e >> 4, c = e & 15;
    __builtin_amdgcn_global_load_async_to_lds_b128(
        (g_v4i*)(Br0 + (size_t)kk * F2P + c), (l_v4i*)(BPr + e), 0, 0);
    __builtin_amdgcn_global_load_async_to_lds_b128(
        (g_v4i*)(Bi0 + (size_t)kk * F2P + c), (l_v4i*)(BPi + e), 0, 0);
  }
  __builtin_amdgcn_s_wait_asynccnt(0);
#else
  for (int e = lane; e < KK * 16; e += 32) {
    int kk = e >> 4, c = e & 15;
    BPr[e] = Br0[(size_t)kk * F2P + c];
    BPi[e] = Bi0[(size_t)kk * F2P + c];
  }
#endif
  __syncthreads();

  v8f cr = {};
  v8f ci = {};
  for (int k0 = 0; k0 < KK; k0 += 4) {
    v2f ar, ai, br, bi;
#pragma unroll
    for (int t = 0; t < 2; ++t) {
      int kk = k0 + khalf + t;
      ar[t] = Ar0[(size_t)kk * BT];
      ai[t] = Ai0[(size_t)kk * BT];
      br[t] = BPr[kk * 16 + ncol];
      bi[t] = BPi[kk * 16 + ncol];
    }
    v2f nbi = -bi;
    // Zr += Ar*Br + Ai*Bi ; Zi += Ai*Br - Ar*Bi   (X * conj(Y))
    cr = __builtin_amdgcn_wmma_f32_16x16x4_f32(false, ar, false, br, (short)0, cr, false, false);
    cr = __builtin_amdgcn_wmma_f32_16x16x4_f32(false, ai, false, bi, (short)0, cr, false, false);
    ci = __builtin_amdgcn_wmma_f32_16x16x4_f32(false, ai, false, br, (short)0, ci, false, false);
    ci = __builtin_amdgcn_wmma_f32_16x16x4_f32(false, ar, false, nbi, (short)0, ci, false, false);
  }
  int mAdd = (lane >> 4) * 8;
#pragma unroll
  for (int r = 0; r < 8; ++r) {
    int brow = bRow0 + mAdd + r;
    size_t zidx = (size_t)p * (BT * F2P) + (size_t)brow * F2P + ocol;
    Zr[zidx] = cr[r];
    Zi[zidx] = ci[r];
  }
}

// Stage-2 synthesis on 12x12, ReLU, quadrature pooling -> feat[b,o]. One block per (b,o).
__global__ __launch_bounds__(192) void k_stage2(const float* __restrict__ DB2, const float* __restrict__ W2J,
                                                const float* __restrict__ Zr, const float* __restrict__ Zi,
                                                float* feat) {
  __shared__ float S2r[144], S2i[144], T2r[144], T2i[144];
  __shared__ float red[192];
  const int b = blockIdx.x, o = blockIdx.y, tid = threadIdx.x;
  const int nth = 192;
  float fsum = 0.f;
  for (int j = 0; j < N2; ++j) {
    for (int e = tid; e < 144; e += nth) { S2r[e] = 0.f; S2i[e] = 0.f; }
    __syncthreads();
    for (int l = 0; l < 6; ++l) {
      int L = 2 * l + 1, off2 = sqoff(l);
      int base = 12 * off2 + j * L * L;
      float fl = (float)(2 * l + 1);
      for (int e = tid; e < L * L; e += nth) {
        int m = e / L, n = e % L;
        float d = DB2[base + e] * fl;
        size_t zidx = (size_t)(off2 + e) * (BT * F2P) + (size_t)b * F2P + o;
        int mi = modp(m - l, N2), ni = modp(n - l, N2);
        S2r[mi * N2 + ni] += d * Zr[zidx];
        S2i[mi * N2 + ni] += d * Zi[zidx];
      }
      __syncthreads();
    }
    for (int e = tid; e < 144; e += nth) {
      int m = e / N2, q = e % N2;
      float sr = 0.f, si = 0.f;
      for (int n = 0; n < N2; ++n) {
        float sv, cv; sincosf((2.0f * PI_F / N2) * (float)(n * q), &sv, &cv);
        float a = S2r[m * N2 + n], bb = S2i[m * N2 + n];
        sr += a * cv - bb * sv;
        si += a * sv + bb * cv;
      }
      T2r[e] = sr; T2i[e] = si;
    }
    __syncthreads();
    float wj = W2J[j];
    for (int e = tid; e < 144; e += nth) {
      int pp = e / N2, q = e % N2;
      float v = 0.f;
      for (int m = 0; m < N2; ++m) {
        float sv, cv; sincosf((2.0f * PI_F / N2) * (float)(m * pp), &sv, &cv);
        v += T2r[m * N2 + q] * cv - T2i[m * N2 + q] * sv;
      }
      fsum += wj * fmaxf(v, 0.f);
    }
    __syncthreads();
  }
  red[tid] = fsum;
  __syncthreads();
  if (tid == 0) {
    float s = 0.f;
    for (int t = 0; t < nth; ++t) s += red[t];
    feat[b * F2c + o] = 0.27415567780803774f * s; // (2*pi/12)^2
  }
}

// out[b,f] = feat[b,:] @ w_out[f,:] + bias[f]  via f32 WMMA (M=128, N=16 pad, K=40)
__global__ __launch_bounds__(32) void k_out(const float* __restrict__ feat, const float* __restrict__ w_out,
                                            const float* __restrict__ bias, float* __restrict__ out) {
  int lane = threadIdx.x;
  int bRow0 = blockIdx.x * 16;
  int arow = bRow0 + (lane & 15);
  int fcol = lane & 15;
  int khalf = (lane >> 4) * 2;
  v8f c = {};
  for (int k0 = 0; k0 < F2c; k0 += 4) {
    v2f a, bm;
#pragma unroll
    for (int t = 0; t < 2; ++t) {
      int kk = k0 + khalf + t;
      a[t] = feat[arow * F2c + kk];
      bm[t] = (fcol < 10) ? w_out[fcol * F2c + kk] : 0.f;
    }
    c = __builtin_amdgcn_wmma_f32_16x16x4_f32(false, a, false, bm, (short)0, c, false, false);
  }
  int mAdd = (lane >> 4) * 8;
  if (fcol < 10) {
    float bv = bias[fcol];
#pragma unroll
    for (int r = 0; r < 8; ++r) {
      int brow = bRow0 + mAdd + r;
      out[brow * 10 + fcol] = c[r] + bv;
    }
  }
}

extern "C" void kernel_launch(void* const* d_in, const int* in_sizes, int n_in,
                              void* d_out, int out_size, void* d_ws, size_t ws_size,
                              hipStream_t stream) {
  (void)in_sizes; (void)n_in; (void)out_size; (void)ws_size;
  const float* x       = (const float*)d_in[0];
  const float* kernel1 = (const float*)d_in[1];
  const float* kernel2 = (const float*)d_in[2];
  const float* w_out   = (const float*)d_in[3];
  const float* bias    = (const float*)d_in[4];
  float* out = (float*)d_out;

  char* base = (char*)d_ws;
  size_t off = 0;
  auto alloc = [&](size_t nfl) -> float* {
    float* p = (float*)(base + off);
    off += nfl * sizeof(float);
    off = (off + 255) & ~(size_t)255;
    return p;
  };
  float* DM1  = alloc((size_t)KT1 * NIN);
  int*   MI1  = (int*)alloc(KT1);
  float* Y1r  = alloc((size_t)KT1 * G1c);
  float* Y1i  = alloc((size_t)KT1 * G1c);
  float* DB1  = alloc((size_t)20 * 1330);
  float* DM2  = alloc((size_t)KT2 * N1);
  int*   MI2  = (int*)alloc(KT2);
  int*   NI2  = (int*)alloc(KT2);
  float* Y2r  = alloc((size_t)KT2 * G3c);
  float* Y2i  = alloc((size_t)KT2 * G3c);
  float* DB2  = alloc((size_t)12 * KT2);
  float* W2J  = alloc(12);
  float* XHr  = alloc((size_t)BT * NIN * NIN);
  float* XHi  = alloc((size_t)BT * NIN * NIN);
  float* Xr   = alloc((size_t)KT1 * BT);
  float* Xi   = alloc((size_t)KT1 * BT);
  float* YK1r = alloc((size_t)KT1 * F1c);
  float* YK1i = alloc((size_t)KT1 * F1c);
  float* X2r  = alloc((size_t)KT2 * F1c * BT);
  float* X2i  = alloc((size_t)KT2 * F1c * BT);
  float* YK2r = alloc((size_t)KT2 * F1c * F2P);
  float* YK2i = alloc((size_t)KT2 * F1c * F2P);
  float* Zr   = alloc((size_t)KT2 * BT * F2P);
  float* Zi   = alloc((size_t)KT2 * BT * F2P);
  float* feat = alloc((size_t)BT * F2c);

  // constant-table setup
  k_dm1<<<dim3(NIN, 10), 256, 0, stream>>>(DM1, MI1);
  k_y1 <<<dim3(3, 10),  256, 0, stream>>>(Y1r, Y1i);
  k_db1<<<dim3(N1, 10), 256, 0, stream>>>(DB1, DM2, MI2, NI2);
  k_db2<<<dim3(N2, 6),  256, 0, stream>>>(DB2, W2J);
  k_y2 <<<dim3(3, 6),   256, 0, stream>>>(Y2r, Y2i);

  // pipeline
  k_fft1  <<<dim3(BT, NIN), 64, 0, stream>>>(x, XHr, XHi);
  k_xproj <<<KT1, 128, 0, stream>>>(DM1, MI1, XHr, XHi, Xr, Xi);
  k_yk1   <<<KT1, 32, 0, stream>>>(Y1r, Y1i, kernel1, YK1r, YK1i);
  k_stage1<<<dim3(BT, F1c), 256, 0, stream>>>(DB1, DM2, MI2, NI2, Xr, Xi, YK1r, YK1i, X2r, X2i);
  k_yk2   <<<KT2, 256, 0, stream>>>(Y2r, Y2i, kernel2, YK2r, YK2i);
  k_zgemm <<<dim3(KT2, BT / 16, 3), 32, 0, stream>>>(X2r, X2i, YK2r, YK2i, Zr, Zi);
  k_stage2<<<dim3(BT, F2c), 192, 0, stream>>>(DB2, W2J, Zr, Zi, feat);
  k_out   <<<BT / 16, 32, 0, stream>>>(feat, w_out, bias, out);
}